// BipartiteMessagePassing_54898271977709
// MI455X (gfx1250) — compile-verified
//
#include <hip/hip_runtime.h>
#include <cstddef>

#define D 128
#define NV 100000
#define NC 50000
#define NE 640000
#define BN_EPS 1e-5f
#define TM 16               // M-tiles (of 16 rows) per block strip

typedef __attribute__((ext_vector_type(2))) float v2f;
typedef __attribute__((ext_vector_type(8))) float v8f;

// ---------------------------------------------------------------- utilities

__global__ void zero_f4_kernel(float4* __restrict__ p, int n4) {
    int i = blockIdx.x * blockDim.x + threadIdx.x;
    if (i < n4) p[i] = make_float4(0.f, 0.f, 0.f, 0.f);
}

// ------------------------------------------------------------- edge scatter
// one wave per edge; lane handles 4 of the 128 dims
__global__ void __launch_bounds__(256) scatter_edges_kernel(
        const float* __restrict__ vfe,
        const float* __restrict__ cfe,
        const int*   __restrict__ eidx,  // [2][NE]
        const float* __restrict__ ew,
        float* __restrict__ vagg,
        float* __restrict__ cagg) {
    int gw = (blockIdx.x * blockDim.x + threadIdx.x) >> 5;   // wave-uniform
    if (gw >= NE) return;
    int lane = threadIdx.x & 31;
    int src = eidx[gw];
    int dst = eidx[NE + gw];
    float w = ew[gw];
    const float4 vf = *(const float4*)(vfe + (size_t)src * D + lane * 4);
    const float4 cf = *(const float4*)(cfe + (size_t)dst * D + lane * 4);
    float* cd = cagg + (size_t)dst * D + lane * 4;
    float* vd = vagg + (size_t)src * D + lane * 4;
    atomicAdd(cd + 0, vf.x * w); atomicAdd(cd + 1, vf.y * w);
    atomicAdd(cd + 2, vf.z * w); atomicAdd(cd + 3, vf.w * w);
    atomicAdd(vd + 0, cf.x * w); atomicAdd(vd + 1, cf.y * w);
    atomicAdd(vd + 2, cf.z * w); atomicAdd(vd + 3, cf.w * w);
}

// ------------------------------------------------- GEMM1 (+ BN statistics)
// h = [feats | agg] @ W1^T + b1 ; per-column sum/sumsq accumulated.
// Flipped WMMA orientation: A = W1 rows (register-resident for the whole
// block strip), B = activation rows streamed per M-tile.
// Two independent accumulation chains (even/odd K) for in-wave ILP.
__global__ void __launch_bounds__(256, 1) gemm1_stats_kernel(
        const float* __restrict__ feats,
        const float* __restrict__ agg,
        const float* __restrict__ W1,   // [D][2D]
        const float* __restrict__ b1,   // [D]
        float* __restrict__ hout,       // [n][D]
        float* __restrict__ gsum,       // [D]
        float* __restrict__ gsq,        // [D]
        int nrows) {
    __shared__ float s_sum[D];
    __shared__ float s_sq[D];
    int tid = threadIdx.x;
    if (tid < D) { s_sum[tid] = 0.f; s_sq[tid] = 0.f; }
    __syncthreads();

    int lane = tid & 31;
    int wave = tid >> 5;
    int l16  = lane & 15;
    int hi   = lane >> 4;      // selects K pair {0,1} vs {2,3} and cols +0/+8
    int koff = hi * 2;
    int n0   = wave * 16;

    // --- preload all 64 weight A-fragments (lane: row n0+l16, K pair) ----
    v2f wA[64];
    const float* wrow = W1 + (size_t)(n0 + l16) * (2 * D) + koff;
#pragma unroll
    for (int kk = 0; kk < 64; ++kk) wA[kk] = *(const v2f*)(wrow + kk * 4);

    v8f bias;
#pragma unroll
    for (int i = 0; i < 8; ++i) bias[i] = b1[n0 + hi * 8 + i];

    float lsum[8], lsq[8];
#pragma unroll
    for (int i = 0; i < 8; ++i) { lsum[i] = 0.f; lsq[i] = 0.f; }

    for (int mt = 0; mt < TM; ++mt) {
        int m0 = (blockIdx.x * TM + mt) * 16;
        if (m0 >= nrows) break;                       // block-uniform guard
        v8f acc0 = bias;                              // even-K chain
        v8f acc1;                                     // odd-K chain
#pragma unroll
        for (int i = 0; i < 8; ++i) acc1[i] = 0.f;

        const float* xr = feats + (size_t)(m0 + l16) * D + koff;
        v2f xb[32];
#pragma unroll
        for (int kk = 0; kk < 32; ++kk) xb[kk] = *(const v2f*)(xr + kk * 4);
#pragma unroll
        for (int j = 0; j < 16; ++j) {
            acc0 = __builtin_amdgcn_wmma_f32_16x16x4_f32(
                false, wA[2 * j],     false, xb[2 * j],     (short)0, acc0, false, false);
            acc1 = __builtin_amdgcn_wmma_f32_16x16x4_f32(
                false, wA[2 * j + 1], false, xb[2 * j + 1], (short)0, acc1, false, false);
        }

        const float* xr2 = agg + (size_t)(m0 + l16) * D + koff;
#pragma unroll
        for (int kk = 0; kk < 32; ++kk) xb[kk] = *(const v2f*)(xr2 + kk * 4);
#pragma unroll
        for (int j = 0; j < 16; ++j) {
            acc0 = __builtin_amdgcn_wmma_f32_16x16x4_f32(
                false, wA[32 + 2 * j],     false, xb[2 * j],     (short)0, acc0, false, false);
            acc1 = __builtin_amdgcn_wmma_f32_16x16x4_f32(
                false, wA[32 + 2 * j + 1], false, xb[2 * j + 1], (short)0, acc1, false, false);
        }

        v8f acc = acc0 + acc1;

        // lane holds row m0+l16, 8 consecutive cols starting at n0+hi*8
        float* hp = hout + (size_t)(m0 + l16) * D + n0 + hi * 8;
        *(float4*)(hp)     = make_float4(acc[0], acc[1], acc[2], acc[3]);
        *(float4*)(hp + 4) = make_float4(acc[4], acc[5], acc[6], acc[7]);
#pragma unroll
        for (int i = 0; i < 8; ++i) {
            float v = acc[i];
            lsum[i] += v;
            lsq[i]  += v * v;
        }
    }

#pragma unroll
    for (int i = 0; i < 8; ++i) {
        int n = n0 + hi * 8 + i;
        atomicAdd(&s_sum[n], lsum[i]);                 // ds_add_f32
        atomicAdd(&s_sq[n],  lsq[i]);
    }
    __syncthreads();
    if (tid < D) {
        atomicAdd(&gsum[tid], s_sum[tid]);
        atomicAdd(&gsq[tid],  s_sq[tid]);
    }
}

// ------------------------------------------------------------- BN folding
__global__ void bn_prep_kernel(const float* __restrict__ stats,  // vs,vsq,cs,csq
                               const float* __restrict__ vg, const float* __restrict__ vbeta,
                               const float* __restrict__ cg, const float* __restrict__ cbeta,
                               float* __restrict__ normv,
                               float* __restrict__ normc) {
    int tid = threadIdx.x;
    if (tid < D) {
        float mu  = stats[tid] * (1.0f / NV);
        float var = stats[D + tid] * (1.0f / NV) - mu * mu;
        float sc  = vg[tid] * rsqrtf(var + BN_EPS);
        normv[tid]     = sc;
        normv[D + tid] = vbeta[tid] - mu * sc;
    } else if (tid < 2 * D) {
        int n = tid - D;
        float mu  = stats[2 * D + n] * (1.0f / NC);
        float var = stats[3 * D + n] * (1.0f / NC) - mu * mu;
        float sc  = cg[n] * rsqrtf(var + BN_EPS);
        normc[n]     = sc;
        normc[D + n] = cbeta[n] - mu * sc;
    }
}

// ---------------------------------------------- GEMM2 (+ residual, output)
// out = feats + relu(h*scale+shift) @ W2^T + b2 (same flipped orientation)
__global__ void __launch_bounds__(256, 1) gemm2_out_kernel(
        const float* __restrict__ hbuf,  // [n][D]
        const float* __restrict__ norm,  // scale,shift
        const float* __restrict__ W2,    // [D][D]
        const float* __restrict__ b2,
        const float* __restrict__ feats,
        float* __restrict__ out,
        int nrows) {
    __shared__ float s_sc[D];
    __shared__ float s_sh[D];
    int tid = threadIdx.x;
    if (tid < D) { s_sc[tid] = norm[tid]; s_sh[tid] = norm[D + tid]; }
    __syncthreads();

    int lane = tid & 31;
    int wave = tid >> 5;
    int l16  = lane & 15;
    int hi   = lane >> 4;
    int koff = hi * 2;
    int n0   = wave * 16;

    v2f wA[32];
    const float* wrow = W2 + (size_t)(n0 + l16) * D + koff;
#pragma unroll
    for (int kk = 0; kk < 32; ++kk) wA[kk] = *(const v2f*)(wrow + kk * 4);

    // normalization coefficients for this lane's K pairs, register-resident
    v2f sc[32], sh[32];
#pragma unroll
    for (int kk = 0; kk < 32; ++kk) {
        int kb = kk * 4 + koff;
        sc[kk].x = s_sc[kb];     sc[kk].y = s_sc[kb + 1];
        sh[kk].x = s_sh[kb];     sh[kk].y = s_sh[kb + 1];
    }

    v8f bias;
#pragma unroll
    for (int i = 0; i < 8; ++i) bias[i] = b2[n0 + hi * 8 + i];

    for (int mt = 0; mt < TM; ++mt) {
        int m0 = (blockIdx.x * TM + mt) * 16;
        if (m0 >= nrows) break;                       // block-uniform guard

        const float* hr = hbuf + (size_t)(m0 + l16) * D + koff;
        v2f xb[32];
#pragma unroll
        for (int kk = 0; kk < 32; ++kk) xb[kk] = *(const v2f*)(hr + kk * 4);
#pragma unroll
        for (int kk = 0; kk < 32; ++kk) {
            v2f a;
            a.x = fmaxf(fmaf(xb[kk].x, sc[kk].x, sh[kk].x), 0.f);
            a.y = fmaxf(fmaf(xb[kk].y, sc[kk].y, sh[kk].y), 0.f);
            xb[kk] = a;
        }
        v8f acc0 = bias;
        v8f acc1;
#pragma unroll
        for (int i = 0; i < 8; ++i) acc1[i] = 0.f;
#pragma unroll
        for (int j = 0; j < 16; ++j) {
            acc0 = __builtin_amdgcn_wmma_f32_16x16x4_f32(
                false, wA[2 * j],     false, xb[2 * j],     (short)0, acc0, false, false);
            acc1 = __builtin_amdgcn_wmma_f32_16x16x4_f32(
                false, wA[2 * j + 1], false, xb[2 * j + 1], (short)0, acc1, false, false);
        }
        v8f acc = acc0 + acc1;

        const float* fp = feats + (size_t)(m0 + l16) * D + n0 + hi * 8;
        float4 f0 = *(const float4*)(fp);
        float4 f1 = *(const float4*)(fp + 4);
        float* op = out + (size_t)(m0 + l16) * D + n0 + hi * 8;
        *(float4*)(op)     = make_float4(acc[0] + f0.x, acc[1] + f0.y,
                                         acc[2] + f0.z, acc[3] + f0.w);
        *(float4*)(op + 4) = make_float4(acc[4] + f1.x, acc[5] + f1.y,
                                         acc[6] + f1.z, acc[7] + f1.w);
    }
}

// ----------------------------------------------------------------- launch

extern "C" void kernel_launch(void* const* d_in, const int* in_sizes, int n_in,
                              void* d_out, int out_size, void* d_ws, size_t ws_size,
                              hipStream_t stream) {
    const float* vfe    = (const float*)d_in[0];
    const float* cfe    = (const float*)d_in[1];
    const int*   eidx   = (const int*)  d_in[2];
    const float* ew     = (const float*)d_in[3];
    const float* vW1    = (const float*)d_in[4];
    const float* vb1    = (const float*)d_in[5];
    const float* vg1    = (const float*)d_in[6];
    const float* vbeta1 = (const float*)d_in[7];
    const float* vW2    = (const float*)d_in[8];
    const float* vb2    = (const float*)d_in[9];
    const float* cW1    = (const float*)d_in[10];
    const float* cb1    = (const float*)d_in[11];
    const float* cg1    = (const float*)d_in[12];
    const float* cbeta1 = (const float*)d_in[13];
    const float* cW2    = (const float*)d_in[14];
    const float* cb2    = (const float*)d_in[15];

    float* ws    = (float*)d_ws;
    float* vagg  = ws;                       // 12,800,000
    float* cagg  = ws + 12800000;            //  6,400,000
    float* hv    = ws + 19200000;            // 12,800,000
    float* hc    = ws + 32000000;            //  6,400,000
    float* stats = ws + 38400000;            // 512: vsum,vsq,csum,csq
    float* normv = ws + 38400512;            // 256
    float* normc = ws + 38400768;            // 256

    float* out_v = (float*)d_out;
    float* out_c = out_v + (size_t)NV * D;

    // 1) zero aggregates + stats
    zero_f4_kernel<<<18750, 256, 0, stream>>>((float4*)ws, 4800000);
    zero_f4_kernel<<<1, 128, 0, stream>>>((float4*)stats, 128);

    // 2) edge scatter (1 wave / edge)
    scatter_edges_kernel<<<NE / 8, 256, 0, stream>>>(vfe, cfe, eidx, ew, vagg, cagg);

    // 3) GEMM1 + BN stats (f32 WMMA). TM*16 = 256 rows per block strip.
    const int gv1 = (NV / 16 + TM - 1) / TM;   // 6250 tiles -> 391 blocks
    const int gc1 = (NC / 16 + TM - 1) / TM;   // 3125 tiles -> 196 blocks
    gemm1_stats_kernel<<<gv1, 256, 0, stream>>>(
        vfe, vagg, vW1, vb1, hv, stats, stats + D, NV);
    gemm1_stats_kernel<<<gc1, 256, 0, stream>>>(
        cfe, cagg, cW1, cb1, hc, stats + 2 * D, stats + 3 * D, NC);

    // 4) fold BN params
    bn_prep_kernel<<<1, 256, 0, stream>>>(stats, vg1, vbeta1, cg1, cbeta1, normv, normc);

    // 5) GEMM2 + residual -> outputs
    gemm2_out_kernel<<<gv1, 256, 0, stream>>>(hv, normv, vW2, vb2, vfe, out_v, NV);
    gemm2_out_kernel<<<gc1, 256, 0, stream>>>(hc, normc, cW2, cb2, cfe, out_c, NC);
}